// DecodeSSDPredictions_73332271612757
// MI455X (gfx1250) — compile-verified
//
#include <hip/hip_runtime.h>

// ---- problem constants (from reference) ----
#define NBOX        8732
#define NFEAT       15          // 3 conf + 4 loc + 4 anchor + 4 var
#define CONF_THR    0.01f
#define IOU_THR     0.45f
#define KNMS        100
#define NPRED       10
#define NBATCH      256
#define NCLS_M1     2           // N_CLASSES-1 foreground classes
#define IMG_SCALE   300.0f

#define NMS_THREADS 512
#define NMS_WAVES   (NMS_THREADS / 32)
#define SPT         18          // ceil(8732/512) boxes per thread

// Per ISA §10.2, a flat pointer into the LDS aperture carries the LDS byte
// offset in its low 32 bits — recover a DS address from a generic pointer.
__device__ __forceinline__ unsigned lds_offset_of(const void* p) {
    return (unsigned)(unsigned long long)p;
}

// gfx1250 async memory->LDS dword copy (ASYNCcnt pipe, no VGPR round-trip).
__device__ __forceinline__ void async_global_to_lds_b32(unsigned lds_addr,
                                                        const void* gaddr) {
    asm volatile("global_load_async_to_lds_b32 %0, %1, off"
                 :: "v"(lds_addr), "v"((unsigned long long)gaddr)
                 : "memory");
}

__device__ __forceinline__ void wait_asynccnt0() {
    asm volatile("s_wait_asynccnt 0x0" ::: "memory");
}

// =====================================================================
// Kernel 1: one workgroup per (batch, class). Decode boxes into
// registers, raw scores async-copied into LDS, run 100 sequential NMS
// rounds with wave32 shuffle argmax + division-free IoU suppression.
// The CONF_THR threshold is folded into the LDS reads, so suppression
// can mark entries with a raw 0.0f.
// =====================================================================
__global__ __launch_bounds__(NMS_THREADS)
void ssd_nms_kernel(const float* __restrict__ y_pred,
                    float* __restrict__ cand)   // [512][100][6]
{
    __shared__ float s_scores[NBOX];      // ~35 KB mutable raw-score vector
    __shared__ float s_rv[NMS_WAVES];
    __shared__ int   s_ri[NMS_WAVES];
    __shared__ float s_bbox[4];
    __shared__ float s_bval;
    __shared__ int   s_bidx;

    const int tid  = threadIdx.x;
    const int b    = blockIdx.x >> 1;     // batch
    const int c    = blockIdx.x & 1;      // foreground class index (0,1)
    const float* __restrict__ base = y_pred + (size_t)b * (NBOX * NFEAT);
    float* __restrict__ wrow = cand + (size_t)blockIdx.x * (KNMS * 6);

    // register-resident decoded boxes for the 18 strided indices this thread owns
    float bx0[SPT], by0[SPT], bx1[SPT], by1[SPT], bar[SPT];

    #pragma unroll
    for (int s = 0; s < SPT; ++s) {
        const int i = tid + s * NMS_THREADS;
        if (i < NBOX) {
            const float* __restrict__ r = base + i * NFEAT;
            // prefetch next strided row for this thread (gfx1250 global_prefetch_b8)
            if (i + NMS_THREADS < NBOX)
                __builtin_prefetch(r + NFEAT * NMS_THREADS, 0, 1);
            // async copy raw conf[..., 1+c] straight into LDS (ASYNCcnt pipe)
            async_global_to_lds_b32(lds_offset_of(&s_scores[i]), r + 1 + c);
            const float l0 = r[3],  l1 = r[4],  l2 = r[5],  l3 = r[6];
            const float a0 = r[7],  a1 = r[8],  a2 = r[9],  a3 = r[10];
            const float v0 = r[11], v1 = r[12], v2 = r[13], v3 = r[14];
            const float cx = l0 * v0 * a2 + a0;
            const float cy = l1 * v1 * a3 + a1;
            const float w  = __expf(l2 * v2) * a2;
            const float h  = __expf(l3 * v3) * a3;
            bx0[s] = (cx - 0.5f * w) * IMG_SCALE;
            by0[s] = (cy - 0.5f * h) * IMG_SCALE;
            bx1[s] = (cx + 0.5f * w) * IMG_SCALE;
            by1[s] = (cy + 0.5f * h) * IMG_SCALE;
            bar[s] = (bx1[s] - bx0[s]) * (by1[s] - by0[s]);
        } else {
            bx0[s] = 0.f; by0[s] = 0.f; bx1[s] = 0.f; by1[s] = 0.f; bar[s] = 0.f;
        }
    }
    wait_asynccnt0();        // all async LDS writes landed
    __syncthreads();

    const int lane = tid & 31;
    const int wv   = tid >> 5;

    for (int k = 0; k < KNMS; ++k) {
        // ---- local argmax over this thread's 18 scores (lowest index on tie);
        //      CONF_THR folded into the read ----
        float mv = -1.0f; int mi = 0x7fffffff;
        #pragma unroll
        for (int s = 0; s < SPT; ++s) {
            const int i = tid + s * NMS_THREADS;
            if (i < NBOX) {
                float v = s_scores[i];
                v = (v > CONF_THR) ? v : 0.0f;
                if (v > mv) { mv = v; mi = i; }  // ascending i -> strict '>' keeps lowest idx
            }
        }
        // ---- wave32 shuffle reduction ----
        #pragma unroll
        for (int off = 16; off > 0; off >>= 1) {
            const float ov = __shfl_xor(mv, off, 32);
            const int   oi = __shfl_xor(mi, off, 32);
            if (ov > mv || (ov == mv && oi < mi)) { mv = ov; mi = oi; }
        }
        if (lane == 0) { s_rv[wv] = mv; s_ri[wv] = mi; }
        __syncthreads();

        // ---- cross-wave reduce + candidate record by thread 0 ----
        if (tid == 0) {
            float bv = s_rv[0]; int bi = s_ri[0];
            #pragma unroll
            for (int w2 = 1; w2 < NMS_WAVES; ++w2) {
                const float ov = s_rv[w2]; const int oi = s_ri[w2];
                if (ov > bv || (ov == bv && oi < bi)) { bv = ov; bi = oi; }
            }
            s_bval = bv; s_bidx = bi;
            if (bv > 0.0f) {
                const float* __restrict__ r = base + bi * NFEAT;   // L0-hot row
                const float l0 = r[3],  l1 = r[4],  l2 = r[5],  l3 = r[6];
                const float a0 = r[7],  a1 = r[8],  a2 = r[9],  a3 = r[10];
                const float v0 = r[11], v1 = r[12], v2 = r[13], v3 = r[14];
                const float cx = l0 * v0 * a2 + a0;
                const float cy = l1 * v1 * a3 + a1;
                const float w  = __expf(l2 * v2) * a2;
                const float h  = __expf(l3 * v3) * a3;
                const float x0 = (cx - 0.5f * w) * IMG_SCALE;
                const float y0 = (cy - 0.5f * h) * IMG_SCALE;
                const float x1 = (cx + 0.5f * w) * IMG_SCALE;
                const float y1 = (cy + 0.5f * h) * IMG_SCALE;
                s_bbox[0] = x0; s_bbox[1] = y0; s_bbox[2] = x1; s_bbox[3] = y1;
                float* o = wrow + k * 6;
                o[0] = (float)(c + 1);  // class id
                o[1] = bv;              // score
                o[2] = x0; o[3] = y0; o[4] = x1; o[5] = y1;
            }
        }
        __syncthreads();

        const float bv = s_bval;                 // uniform across block
        if (bv <= 0.0f) {                        // all remaining rounds yield zero rows
            for (int j = k * 6 + tid; j < KNMS * 6; j += NMS_THREADS)
                wrow[j] = 0.0f;
            break;
        }
        const int   bi  = s_bidx;
        const float qx0 = s_bbox[0], qy0 = s_bbox[1], qx1 = s_bbox[2], qy1 = s_bbox[3];
        const float qar = (qx1 - qx0) * (qy1 - qy0);

        // ---- suppression: division-free IoU >= T  <=>  inter >= T*(a1+a2-inter+eps) ----
        #pragma unroll
        for (int s = 0; s < SPT; ++s) {
            const int i = tid + s * NMS_THREADS;
            if (i < NBOX) {
                float ix = fminf(qx1, bx1[s]) - fmaxf(qx0, bx0[s]);
                float iy = fminf(qy1, by1[s]) - fmaxf(qy0, by0[s]);
                ix = fmaxf(ix, 0.0f); iy = fmaxf(iy, 0.0f);
                const float inter = ix * iy;
                const float denom = qar + bar[s] - inter + 1e-8f;
                if (inter >= IOU_THR * denom || i == bi)
                    s_scores[i] = 0.0f;          // raw 0 -> effective 0 after threshold
            }
        }
        __syncthreads();
    }
}

// =====================================================================
// Kernel 2: one workgroup per batch; top-10 of 200 candidate scores
// (sequential stable argmax reproduces jax.lax.top_k tie ordering).
// =====================================================================
#define TOPK_THREADS 256

__global__ __launch_bounds__(TOPK_THREADS)
void ssd_topk_kernel(const float* __restrict__ cand,  // [256][200][6]
                     float* __restrict__ out)         // [256][10][6]
{
    __shared__ float sv[TOPK_THREADS];
    __shared__ int   si[TOPK_THREADS];

    const int tid = threadIdx.x;
    const int b   = blockIdx.x;
    const float* __restrict__ cb = cand + (size_t)b * (NCLS_M1 * KNMS * 6);

    float myv = (tid < NCLS_M1 * KNMS) ? cb[tid * 6 + 1] : -1.0f;

    for (int j = 0; j < NPRED; ++j) {
        sv[tid] = myv; si[tid] = tid;
        __syncthreads();
        #pragma unroll
        for (int st = TOPK_THREADS / 2; st > 0; st >>= 1) {
            if (tid < st) {
                const float ov = sv[tid + st]; const int oi = si[tid + st];
                if (ov > sv[tid] || (ov == sv[tid] && oi < si[tid])) {
                    sv[tid] = ov; si[tid] = oi;
                }
            }
            __syncthreads();
        }
        const int bi = si[0];          // read before anyone rewrites sv/si
        if (tid == bi) myv = -1.0f;    // remove picked candidate
        if (tid < 6)
            out[(size_t)b * (NPRED * 6) + j * 6 + tid] = cb[bi * 6 + tid];
        __syncthreads();
    }
}

// =====================================================================
extern "C" void kernel_launch(void* const* d_in, const int* in_sizes, int n_in,
                              void* d_out, int out_size, void* d_ws, size_t ws_size,
                              hipStream_t stream)
{
    const float* y_pred = (const float*)d_in[0];
    float* out  = (float*)d_out;
    float* cand = (float*)d_ws;   // needs 512*100*6*4 = 1.2 MB

    ssd_nms_kernel<<<NBATCH * NCLS_M1, NMS_THREADS, 0, stream>>>(y_pred, cand);
    ssd_topk_kernel<<<NBATCH, TOPK_THREADS, 0, stream>>>(cand, out);
}